// PCNetwork_16647293239515
// MI455X (gfx1250) — compile-verified
//
#include <hip/hip_runtime.h>
#include <hip/hip_bf16.h>

typedef __attribute__((ext_vector_type(2))) float v2f;
typedef __attribute__((ext_vector_type(4))) float v4f;
typedef __attribute__((ext_vector_type(8))) float v8f;
typedef __attribute__((ext_vector_type(4))) int   v4i;

// CDNA5 async global->LDS path (ASYNCcnt-tracked), if this toolchain exposes it.
#if __has_builtin(__builtin_amdgcn_global_load_async_to_lds_b128) && \
    __has_builtin(__builtin_amdgcn_s_wait_asynccnt)
#define PC_ASYNC_LDS 1
#else
#define PC_ASYNC_LDS 0
#endif

// Builtin wants typed int4 pointers: (v4i AS1*, v4i AS3*, imm offset, imm cpol).
#define PC_GPTR(p) ((__attribute__((address_space(1))) v4i*)(p))
#define PC_LPTR(p) ((__attribute__((address_space(3))) v4i*)(p))

#define MUE_WAVES 4
#define MUE_ROWS  (MUE_WAVES * 16)   // 64 rows per block (16 rows per wave)
#define MUE_KT    64                 // K tile in floats (16 KB per buffer)
#define MUE_PAD   68                 // row stride: 32-lane ds_load_b64 hits all 64 banks once

// ---------------------------------------------------------------------------
// e[r] = xprev[r] - tanh( dot(W[r,:], xin) ),  R rows of length C.
// W tiles are staged into LDS (async copy when available, double-buffered),
// then fed to V_WMMA_F32_16X16X4_F32:
//   A (16x4 f32): lane = M (lane&15), lane-half selects K pair (from LDS).
//   B (4x16 f32): xin in column N=0 only (lanes 0 and 16).
//   D column 0 -> lanes 0 (M=0..7 in v0..7) and 16 (M=8..15).
// Grids are exact (R % 64 == 0, C % 64 == 0) so barriers are uniform.
// ---------------------------------------------------------------------------
__global__ void __launch_bounds__(MUE_WAVES * 32)
pc_mu_e(const float* __restrict__ W, const float* __restrict__ xin,
        const float* __restrict__ xprev, float* __restrict__ e,
        int R, int C) {
  __shared__ float tile[2][MUE_ROWS][MUE_PAD];

  const int tid   = threadIdx.x;
  const int lane  = tid & 31;
  const int wave  = tid >> 5;
  const int mrow  = lane & 15;
  const int khalf = lane >> 4;
  const int rBlk  = blockIdx.x * MUE_ROWS;

  // Copy mapping: each thread moves 8 x 16B chunks per tile (64x64 floats).
  const int crow0 = tid >> 4;          // 0..7
  const int ccol  = (tid & 15) * 4;    // 0,4,...,60
  const int nt    = C / MUE_KT;

  const float sel = (mrow == 0) ? 1.0f : 0.0f;   // only column 0 of B is live

  // Prologue: issue copy of tile 0 into buffer 0.
  {
    const float* gbase = W + (size_t)rBlk * (size_t)C;
#pragma unroll
    for (int i = 0; i < 8; ++i) {
      const int row = crow0 + i * 8;
      const float* g = gbase + (size_t)row * (size_t)C + ccol;
      float* l = &tile[0][row][ccol];
#if PC_ASYNC_LDS
      __builtin_amdgcn_global_load_async_to_lds_b128(PC_GPTR(g), PC_LPTR(l), 0, 0);
#else
      *(v4f*)l = *(const v4f*)g;
#endif
    }
  }

  v8f acc = {0.f, 0.f, 0.f, 0.f, 0.f, 0.f, 0.f, 0.f};
  for (int t = 0; t < nt; ++t) {
    const int cur = t & 1;
#if PC_ASYNC_LDS
    __builtin_amdgcn_s_wait_asynccnt(0);   // copies into tile[cur] landed
#endif
    __syncthreads();                       // ..for every wave; tile[1-cur] free

    if (t + 1 < nt) {                      // issue next tile into other buffer
      const float* gbase =
          W + (size_t)rBlk * (size_t)C + (size_t)(t + 1) * MUE_KT;
#pragma unroll
      for (int i = 0; i < 8; ++i) {
        const int row = crow0 + i * 8;
        const float* g = gbase + (size_t)row * (size_t)C + ccol;
        float* l = &tile[1 - cur][row][ccol];
#if PC_ASYNC_LDS
        __builtin_amdgcn_global_load_async_to_lds_b128(PC_GPTR(g), PC_LPTR(l), 0, 0);
#else
        *(v4f*)l = *(const v4f*)g;
#endif
      }
    }

    const int k0 = t * MUE_KT;
#pragma unroll 4
    for (int kk = 0; kk < MUE_KT; kk += 4) {
      v2f a = *(const v2f*)&tile[cur][wave * 16 + mrow][kk + 2 * khalf];
      v2f b;
      b.x = sel * xin[k0 + kk + 2 * khalf];
      b.y = sel * xin[k0 + kk + 2 * khalf + 1];
      acc = __builtin_amdgcn_wmma_f32_16x16x4_f32(false, a, false, b,
                                                  (short)0, acc, false, false);
    }
  }

  if (mrow == 0) {                         // lanes 0 and 16 hold column N=0
    const int base = rBlk + wave * 16 + khalf * 8;
#pragma unroll
    for (int j = 0; j < 8; ++j)
      e[base + j] = xprev[base + j] - tanhf(acc[j]);
  }
}

// ---------------------------------------------------------------------------
// grad[c] = sum_r e[r] * W[r,c]  (c in [0,C), reduce over R rows).
// A = e in matrix row M=0, B = W 4x16 tile (16-lane x 4B contiguous per row).
// Runs right after pc_mu_e on the same matrix -> mostly L2 hits (128MiB < 192MB).
// ---------------------------------------------------------------------------
__global__ void pc_grad(const float* __restrict__ W,
                        const float* __restrict__ e,
                        float* __restrict__ grad,
                        int R, int C) {
  const int lane  = threadIdx.x & 31;
  const int wave  = threadIdx.x >> 5;
  const int n     = lane & 15;
  const int khalf = lane >> 4;
  const int c0    = (blockIdx.x * (blockDim.x >> 5) + wave) * 16;
  if (c0 >= C) return;

  const float* wcol = W + (size_t)(2 * khalf) * (size_t)C + (size_t)(c0 + n);
  const float  sel  = (n == 0) ? 1.0f : 0.0f;   // only matrix row M=0 of A is live

  v8f acc = {0.f, 0.f, 0.f, 0.f, 0.f, 0.f, 0.f, 0.f};
  for (int k0 = 0; k0 < R; k0 += 4) {
    v2f b;
    b.x = wcol[(size_t)k0 * (size_t)C];          // W[k0 + 2*khalf    ][c0+n]
    b.y = wcol[(size_t)(k0 + 1) * (size_t)C];    // W[k0 + 2*khalf + 1][c0+n]
    v2f a;
    a.x = sel * e[k0 + 2 * khalf];               // A[0][k] = e[k0+k]
    a.y = sel * e[k0 + 2 * khalf + 1];
    __builtin_prefetch(wcol + (size_t)(k0 + 32) * (size_t)C, 0, 3);
    acc = __builtin_amdgcn_wmma_f32_16x16x4_f32(false, a, false, b,
                                                (short)0, acc, false, false);
  }

  if (lane < 16) grad[c0 + n] = acc[0];          // D[0][n]: lanes 0..15, v0
}

// ---------------------------------------------------------------------------
// State update + deterministic per-block error partials.
// ---------------------------------------------------------------------------
__global__ void pc_update(float* __restrict__ x1, float* __restrict__ x2,
                          const float* __restrict__ e0, const float* __restrict__ e1,
                          const float* __restrict__ g1, const float* __restrict__ g2,
                          float* __restrict__ partials, int n1, int n2) {
  const int i = blockIdx.x * blockDim.x + threadIdx.x;
  float err = 0.f;
  if (i < n2) {
    float e0v = e0[i];
    err += e0v * e0v;
    float x2v = x2[i];                     // e2 = x2_old (top-layer mu is zero)
    err += x2v * x2v;
    float g = -x2v + g2[i];
    g = fminf(1.f, fmaxf(-1.f, g));
    x2[i] = tanhf(x2v + 0.01f * g);
  }
  if (i < n1) {
    float e1v = e1[i];
    err += e1v * e1v;
    float g = -e1v + g1[i];
    g = fminf(1.f, fmaxf(-1.f, g));
    x1[i] = tanhf(x1[i] + 0.01f * g);
  }
  __shared__ float red[256];
  red[threadIdx.x] = err;
  __syncthreads();
  for (int s = 128; s > 0; s >>= 1) {
    if ((int)threadIdx.x < s) red[threadIdx.x] += red[threadIdx.x + s];
    __syncthreads();
  }
  if (threadIdx.x == 0) partials[blockIdx.x] = red[0];
}

__global__ void pc_zero(float* __restrict__ p, int n) {
  int i = blockIdx.x * blockDim.x + threadIdx.x;
  if (i < n) p[i] = 0.f;
}

// Fixed-order final sum -> deterministic scalar output.
__global__ void pc_final(const float* __restrict__ partials, float* __restrict__ out,
                         int n) {
  if (threadIdx.x == 0 && blockIdx.x == 0) {
    float s = 0.f;
    for (int i = 0; i < n; ++i) s += partials[i];
    out[0] = s;
  }
}

extern "C" void kernel_launch(void* const* d_in, const int* in_sizes, int n_in,
                              void* d_out, int out_size, void* d_ws, size_t ws_size,
                              hipStream_t stream) {
  const float* x0 = (const float*)d_in[0];   // input_data [n0]
  const float* W0 = (const float*)d_in[1];   // [n0, n1]
  const float* W1 = (const float*)d_in[2];   // [n1, n2]
  const int STEPS = 10;                      // reference setup constant (fixed launch
                                             // sequence required for graph capture)

  const int n0 = in_sizes[0];                // 8192
  const int n1 = in_sizes[1] / n0;           // 4096
  const int n2 = in_sizes[2] / n1;           // 8192

  // Workspace layout (floats)
  float* w  = (float*)d_ws;
  float* x1 = w;                 // n1
  float* x2 = x1 + n1;           // n2
  float* e0 = x2 + n2;           // n0
  float* e1 = e0 + n0;           // n1
  float* g1 = e1 + n1;           // n1
  float* g2 = g1 + n1;           // n2
  float* pp = g2 + n2;           // partials (<=64)

  const int upBlocks = (n2 + 255) / 256;     // 32 partials

  // Fresh state each call (ws is not re-poisoned between replays).
  pc_zero<<<(n1 + n2 + 255) / 256, 256, 0, stream>>>(x1, n1 + n2);

  for (int s = 0; s < STEPS; ++s) {
    // Pass W0 twice back-to-back (second pass ~L2 hits), then W1 twice.
    pc_mu_e<<<n0 / MUE_ROWS, MUE_WAVES * 32, 0, stream>>>(W0, x1, x0, e0, n0, n1);
    pc_grad<<<n1 / 64, 128, 0, stream>>>(W0, e0, g1, n0, n1);
    pc_mu_e<<<n1 / MUE_ROWS, MUE_WAVES * 32, 0, stream>>>(W1, x2, x1, e1, n1, n2);
    pc_grad<<<n2 / 64, 128, 0, stream>>>(W1, e1, g2, n1, n2);
    // State update + error partials (final step's partials == errs[-1]).
    pc_update<<<upBlocks, 256, 0, stream>>>(x1, x2, e0, e1, g1, g2, pp, n1, n2);
  }

  pc_final<<<1, 32, 0, stream>>>(pp, (float*)d_out, upBlocks);
}